// MLPCritic_27230092657235
// MI455X (gfx1250) — compile-verified
//
#include <hip/hip_runtime.h>
#include <hip/hip_bf16.h>
#include <math.h>

// ---------------------------------------------------------------------------
// MLPCritic forward for MI455X (gfx1250, wave32).
// Memory-bound pipeline (~435 MB/invocation); fp32 throughout to match
// reference numerics exactly.
//   - adj_v GEMMs (M=2048,K=2048,N=16): V_WMMA_F32_16X16X4_F32 matrix cores,
//     with the B operand staged to LDS via GLOBAL_LOAD_ASYNC_TO_LDS_B128.
//   - dominant obs@Wm1 stage (318 MB): coalesced Wm1 streaming with the obs
//     chunk async-staged to LDS once per block.
// ---------------------------------------------------------------------------

typedef __attribute__((ext_vector_type(2))) float v2f;
typedef __attribute__((ext_vector_type(8))) float v8f;

static constexpr int N_NODES = 2048;
static constexpr int N_EDGES = 4096;
static constexpr int L       = 4096;
static constexpr int H1      = 1024;
static constexpr int DIN     = L * 19;      // 77824
static constexpr int CHUNK   = 1024;        // DIN = 76 * 1024
static constexpr int NCHUNK  = DIN / CHUNK; // 76

// Issue one async global->LDS 128-bit copy. lds_byte is the byte offset of the
// destination inside this kernel's (single) static LDS allocation; gsrc is a
// 16B-aligned global source. Tracked by ASYNCcnt.
__device__ __forceinline__ void async_copy_b128(unsigned lds_byte,
                                                const float* gsrc) {
    asm volatile("global_load_async_to_lds_b128 %0, %1, off"
                 :: "v"(lds_byte), "v"(gsrc)
                 : "memory");
}
__device__ __forceinline__ void async_wait0() {
    asm volatile("s_wait_asynccnt 0x0" ::: "memory");
}

// ---------------- generic row-parallel matvec: y = A[rows,cols] @ x --------
__global__ void matvec_kernel(const float* __restrict__ A,
                              const float* __restrict__ x,
                              float* __restrict__ y, int cols) {
    const int row = blockIdx.x;
    const float* a = A + (size_t)row * cols;
    float s = 0.f;
    for (int i = threadIdx.x; i < cols; i += blockDim.x) s += a[i] * x[i];
    __shared__ float red[256];
    red[threadIdx.x] = s;
    __syncthreads();
    for (int off = 128; off > 0; off >>= 1) {
        if (threadIdx.x < off) red[threadIdx.x] += red[threadIdx.x + off];
        __syncthreads();
    }
    if (threadIdx.x == 0) y[row] = red[0];
}

// ---------------- G1 = concat(X, TZe) @ W1  ([2048,2] @ [2,16]) ------------
__global__ void h0w1_kernel(const float* __restrict__ X,
                            const float* __restrict__ TZe,
                            const float* __restrict__ W1,
                            float* __restrict__ G1) {
    int i = blockIdx.x * blockDim.x + threadIdx.x;
    if (i >= N_NODES * 16) return;
    int r = i >> 4, j = i & 15;
    G1[i] = X[r] * W1[j] + TZe[r] * W1[16 + j];
}

// ---------------- Out = act(adj_v @ B + bias): WMMA f32 16x16x4 ------------
// One wave computes one 16x16 output tile. M = 2048 -> 128 waves exactly, so
// every wave is fully active (EXEC all-1s as WMMA requires).
// B ([2048,16] = 128KB) is async-staged into LDS once per block; every wave
// then reads it as LDS broadcasts instead of redundant global traffic.
// A-matrix 16x4 f32 layout: lanes 0-15 hold K=0,1; lanes 16-31 hold K=2,3.
// B 4x16: lane n holds column n; lanes 0-15 rows 0,1; lanes 16-31 rows 2,3.
// D: VGPR j, lanes 0-15 -> M=j, lanes 16-31 -> M=8+j; N = lane%16.
__global__ void gemm_adjv_wmma(const float* __restrict__ A,   // [2048,2048]
                               const float* __restrict__ B,   // [2048,16]
                               const float* __restrict__ bias,// [16]
                               float* __restrict__ Out,       // [2048,16]
                               int doRelu) {
    __shared__ float sB[N_NODES * 16];                 // 128 KB (only LDS obj)

    // async stage: 256 threads x 32 rounds x 16B = 128 KB
    {
        const int t = threadIdx.x;
        #pragma unroll 4
        for (int r = 0; r < 32; ++r) {
            const int fidx = r * 1024 + t * 4;         // float index
            async_copy_b128((unsigned)(fidx * 4), B + fidx);
        }
        async_wait0();
        __syncthreads();
    }

    const int wave = (blockIdx.x * blockDim.x + threadIdx.x) >> 5;
    const int lane = threadIdx.x & 31;
    const int half = lane >> 4;    // 0: K pair {0,1}; 1: K pair {2,3}
    const int l16  = lane & 15;
    const int row0 = wave * 16;

    v8f c = {0.f, 0.f, 0.f, 0.f, 0.f, 0.f, 0.f, 0.f};
    const float* Arow = A + (size_t)(row0 + l16) * N_NODES + 2 * half;
    const float* Bl   = sB + (2 * half) * 16 + l16;

    for (int k0 = 0; k0 < N_NODES; k0 += 4) {
        v2f a, b;
        a.x = Arow[k0];
        a.y = Arow[k0 + 1];
        b.x = Bl[k0 * 16];
        b.y = Bl[k0 * 16 + 16];
        c = __builtin_amdgcn_wmma_f32_16x16x4_f32(
                /*neg_a=*/false, a, /*neg_b=*/false, b,
                /*c_mod=*/(short)0, c, /*reuse_a=*/false, /*reuse_b=*/false);
    }

    const float bj = bias[l16];
    #pragma unroll
    for (int j = 0; j < 8; ++j) {
        int m = row0 + j + 8 * half;
        float v = c[j] + bj;
        if (doRelu) v = fmaxf(v, 0.f);
        Out[(size_t)m * 16 + l16] = v;
    }
}

// ---------------- G2 = Hh @ W2 ([2048,16] @ [16,16], no bias) --------------
__global__ void gemm16_kernel(const float* __restrict__ Hh,
                              const float* __restrict__ W2,
                              float* __restrict__ G2) {
    int i = blockIdx.x * blockDim.x + threadIdx.x;
    if (i >= N_NODES * 16) return;
    int r = i >> 4, j = i & 15;
    const float* h = Hh + (size_t)r * 16;
    float s = 0.f;
    #pragma unroll
    for (int k = 0; k < 16; ++k) s += h[k] * W2[k * 16 + j];
    G2[i] = s;
}

// ---------------- logsumexp(cost) + prefix-sum of flag (single block) ------
__global__ void prep_kernel(const float* __restrict__ cost,
                            const float* __restrict__ flag,
                            int* __restrict__ src,
                            float* __restrict__ lse) {
    __shared__ float sred[1024];
    __shared__ int   sscan[1024];
    const int t = threadIdx.x;

    // max over cost
    float m = -INFINITY;
    for (int i = t; i < L; i += 1024) m = fmaxf(m, cost[i]);
    sred[t] = m; __syncthreads();
    for (int off = 512; off > 0; off >>= 1) {
        if (t < off) sred[t] = fmaxf(sred[t], sred[t + off]);
        __syncthreads();
    }
    const float mx = sred[0];
    __syncthreads();

    // sum exp(cost - max)
    float s = 0.f;
    for (int i = t; i < L; i += 1024) s += expf(cost[i] - mx);
    sred[t] = s; __syncthreads();
    for (int off = 512; off > 0; off >>= 1) {
        if (t < off) sred[t] += sred[t + off];
        __syncthreads();
    }
    if (t == 0) *lse = mx + logf(sred[0]);

    // inclusive prefix sum of flag (4 elems per thread, Hillis-Steele across)
    const int base = t * 4;
    int f0 = (int)flag[base + 0], f1 = (int)flag[base + 1];
    int f2 = (int)flag[base + 2], f3 = (int)flag[base + 3];
    const int tot = f0 + f1 + f2 + f3;
    sscan[t] = tot; __syncthreads();
    for (int off = 1; off < 1024; off <<= 1) {
        int v = sscan[t];
        int add = (t >= off) ? sscan[t - off] : 0;
        __syncthreads();
        sscan[t] = v + add;
        __syncthreads();
    }
    int run = sscan[t] - tot;   // exclusive prefix
    int c0 = run + f0, c1 = c0 + f1, c2 = c1 + f2, c3 = c2 + f3;
    src[base + 0] = min(max(c0 - 1, 0), N_NODES - 1);
    src[base + 1] = min(max(c1 - 1, 0), N_NODES - 1);
    src[base + 2] = min(max(c2 - 1, 0), N_NODES - 1);
    src[base + 3] = min(max(c3 - 1, 0), N_NODES - 1);
}

// ---------------- obs assembly: [L,19] row-major flattened ------------------
__global__ void obs_kernel(const float* __restrict__ cost,
                           const float* __restrict__ flag,
                           const float* __restrict__ dual,
                           const float* __restrict__ emb,
                           const int* __restrict__ src,
                           const float* __restrict__ lse,
                           float* __restrict__ obs) {
    int i = blockIdx.x * blockDim.x + threadIdx.x;
    if (i >= L) return;
    const float l = *lse;
    float* o = obs + (size_t)i * 19;
    const float f = flag[i];
    const bool on = f > 0.f;
    const int sidx = src[i];
    o[0] = cost[i] - l;
    o[1] = f;
    o[2] = on ? dual[sidx] : 1.0f;
    const float* e = emb + (size_t)sidx * 16;
    #pragma unroll
    for (int j = 0; j < 16; ++j) o[3 + j] = on ? e[j] : 0.f;
}

// ---------------- big layer: partials of obs @ Wm1 (coalesced over j) -------
// grid (H1/256, NCHUNK); the 4KB obs chunk is async-staged to LDS (one B128
// per thread), so the inner loop is pure coalesced Wm1 streaming.
__global__ void mlp1_partial_kernel(const float* __restrict__ obs,
                                    const float* __restrict__ Wm1,
                                    float* __restrict__ part) {
    __shared__ float s_obs[CHUNK];                     // 4 KB (only LDS obj)
    const int j = blockIdx.x * 256 + threadIdx.x;      // output column
    const int c = blockIdx.y;                          // row chunk
    const int i0 = c * CHUNK;

    async_copy_b128((unsigned)(threadIdx.x * 16), obs + i0 + threadIdx.x * 4);
    async_wait0();
    __syncthreads();

    const float* w = Wm1 + (size_t)i0 * H1 + j;
    float s = 0.f;
    for (int i = 0; i < CHUNK; ++i) s += s_obs[i] * w[(size_t)i * H1];
    part[(size_t)c * H1 + j] = s;
}

__global__ void mlp1_reduce_kernel(const float* __restrict__ part,
                                   const float* __restrict__ bm1,
                                   float* __restrict__ h1) {
    int j = blockIdx.x * blockDim.x + threadIdx.x;
    if (j >= H1) return;
    float s = 0.f;
    for (int c = 0; c < NCHUNK; ++c) s += part[(size_t)c * H1 + j];
    h1[j] = tanhf(s + bm1[j]);
}

// ---------------- layer 2: h2 = tanh(h1 @ Wm2 + bm2) -----------------------
__global__ void mlp2_kernel(const float* __restrict__ h1,
                            const float* __restrict__ Wm2,
                            const float* __restrict__ bm2,
                            float* __restrict__ h2) {
    int j = blockIdx.x * blockDim.x + threadIdx.x;
    if (j >= H1) return;
    float s = 0.f;
    for (int i = 0; i < H1; ++i) s += h1[i] * Wm2[(size_t)i * H1 + j];
    h2[j] = tanhf(s + bm2[j]);
}

// ---------------- layer 3: scalar v = h2 @ Wm3 + bm3 -----------------------
__global__ void mlp3_kernel(const float* __restrict__ h2,
                            const float* __restrict__ Wm3,
                            const float* __restrict__ bm3,
                            float* __restrict__ out) {
    __shared__ float red[1024];
    const int t = threadIdx.x;
    red[t] = h2[t] * Wm3[t];
    __syncthreads();
    for (int off = 512; off > 0; off >>= 1) {
        if (t < off) red[t] += red[t + off];
        __syncthreads();
    }
    if (t == 0) out[0] = red[0] + bm3[0];
}

extern "C" void kernel_launch(void* const* d_in, const int* in_sizes, int n_in,
                              void* d_out, int out_size, void* d_ws, size_t ws_size,
                              hipStream_t stream) {
    const float* X          = (const float*)d_in[0];
    const float* Z          = (const float*)d_in[1];
    const float* adj_e      = (const float*)d_in[2];
    const float* adj_v      = (const float*)d_in[3];
    const float* T          = (const float*)d_in[4];
    const float* state_cost = (const float*)d_in[5];
    const float* state_flag = (const float*)d_in[6];
    const float* state_dual = (const float*)d_in[7];
    const float* W1         = (const float*)d_in[8];
    const float* b1         = (const float*)d_in[9];
    const float* W2         = (const float*)d_in[10];
    const float* b2         = (const float*)d_in[11];
    const float* Wm1        = (const float*)d_in[12];
    const float* bm1        = (const float*)d_in[13];
    const float* Wm2        = (const float*)d_in[14];
    const float* bm2        = (const float*)d_in[15];
    const float* Wm3        = (const float*)d_in[16];
    const float* bm3        = (const float*)d_in[17];
    float* out = (float*)d_out;

    // workspace carve-up (fp32)
    float* ws   = (float*)d_ws;
    float* Ze   = ws;                 // 4096
    float* TZe  = Ze + N_EDGES;       // 2048
    float* G1   = TZe + N_NODES;      // 32768
    float* Hh   = G1 + N_NODES * 16;  // 32768
    float* G2   = Hh + N_NODES * 16;  // 32768
    float* emb  = G2 + N_NODES * 16;  // 32768
    float* lse  = emb + N_NODES * 16; // 1
    int*   src  = (int*)(lse + 1);    // 4096 ints
    float* obs  = (float*)(src + L);  // 77824
    float* part = obs + DIN;          // 76*1024
    float* h1   = part + NCHUNK * H1; // 1024
    float* h2   = h1 + H1;            // 1024

    // 1) Ze = adj_e @ Z                       (64 MB, memory bound)
    matvec_kernel<<<N_EDGES, 256, 0, stream>>>(adj_e, Z, Ze, N_EDGES);
    // 2) TZe = T @ Ze                         (32 MB)
    matvec_kernel<<<N_NODES, 256, 0, stream>>>(T, Ze, TZe, N_EDGES);
    // 3) G1 = [X, TZe] @ W1
    h0w1_kernel<<<(N_NODES * 16 + 255) / 256, 256, 0, stream>>>(X, TZe, W1, G1);
    // 4) Hh = relu(adj_v @ G1 + b1)           (WMMA f32, 128 tiles)
    gemm_adjv_wmma<<<16, 256, 0, stream>>>(adj_v, G1, b1, Hh, 1);
    // 5) G2 = Hh @ W2
    gemm16_kernel<<<(N_NODES * 16 + 255) / 256, 256, 0, stream>>>(Hh, W2, G2);
    // 6) emb = adj_v @ G2 + b2                (WMMA f32)
    gemm_adjv_wmma<<<16, 256, 0, stream>>>(adj_v, G2, b2, emb, 0);
    // 7) logsumexp + flag prefix-sum, then obs assembly
    prep_kernel<<<1, 1024, 0, stream>>>(state_cost, state_flag, src, lse);
    obs_kernel<<<(L + 255) / 256, 256, 0, stream>>>(state_cost, state_flag,
                                                    state_dual, emb, src, lse, obs);
    // 8) h1 = tanh(obs @ Wm1 + bm1)           (318 MB — dominant stage)
    {
        dim3 grid(H1 / 256, NCHUNK);
        mlp1_partial_kernel<<<grid, 256, 0, stream>>>(obs, Wm1, part);
        mlp1_reduce_kernel<<<(H1 + 255) / 256, 256, 0, stream>>>(part, bm1, h1);
    }
    // 9) h2 = tanh(h1 @ Wm2 + bm2)
    mlp2_kernel<<<(H1 + 255) / 256, 256, 0, stream>>>(h1, Wm2, bm2, h2);
    // 10) v = h2 @ Wm3 + bm3
    mlp3_kernel<<<1, 1024, 0, stream>>>(h2, Wm3, bm3, out);
}